// GCN_38001870635091
// MI455X (gfx1250) — compile-verified
//
#include <hip/hip_runtime.h>

#define D 128
#define WSTR 132   // padded LDS row stride (floats): 132 % 64 = 4 -> conflict-free column reads

typedef __attribute__((ext_vector_type(2))) float v2f;
typedef __attribute__((ext_vector_type(8))) float v8f;

// ---------------------------------------------------------------------------
// Utility: zero a float buffer (grid-stride)
// ---------------------------------------------------------------------------
__global__ void gcn_zero_f32(float* __restrict__ p, long long n) {
  long long i = (long long)blockIdx.x * blockDim.x + threadIdx.x;
  long long stride = (long long)gridDim.x * blockDim.x;
  for (; i < n; i += stride) p[i] = 0.0f;
}

// ---------------------------------------------------------------------------
// Degree accumulation: out_deg[src[e]] += 1 ; in_deg[dst[e]] += 1
// ---------------------------------------------------------------------------
__global__ void gcn_degree(const int* __restrict__ src, const int* __restrict__ dst,
                           float* __restrict__ odeg, float* __restrict__ ideg, int E) {
  int i = blockIdx.x * blockDim.x + threadIdx.x;
  if (i < E) {
    atomicAdd(&odeg[src[i]], 1.0f);
    atomicAdd(&ideg[dst[i]], 1.0f);
  }
}

// deg -> clip(deg,1)^-0.5, in place
__global__ void gcn_norm(float* __restrict__ x, int n) {
  int i = blockIdx.x * blockDim.x + threadIdx.x;
  if (i < n) x[i] = rsqrtf(fmaxf(x[i], 1.0f));
}

// ---------------------------------------------------------------------------
// hs = (h @ W^T + b) * out_norm[:,None]   via V_WMMA_F32_16X16X4_F32
//
// Block: 128 threads = 4 waves. blockIdx.y in {0,1} selects a 64-column group.
// Wave w computes output columns [nbase + 16w, nbase + 16w + 16) for one
// 16-row tile; blocks stride over row tiles.
//
// Fragment layouts (ISA 7.12.2, 32-bit 16x4 A / 16x16 C):
//   A: lanes 0-15 hold M=lane,  VGPR0=K+0, VGPR1=K+1 ; lanes 16-31: K+2, K+3
//   B: lanes 0-15 hold N=lane,  VGPR0=K+0, VGPR1=K+1 ; lanes 16-31: K+2, K+3
//      (B[k][n] = W[n][k] since hs = h @ W^T)
//   C: VGPR r -> M = r + 8*(lane>=16), N = lane & 15
// ---------------------------------------------------------------------------
__global__ __launch_bounds__(128)
void gcn_gemm_wmma(const float* __restrict__ h, const float* __restrict__ W,
                   const float* __restrict__ bias, const float* __restrict__ out_norm,
                   float* __restrict__ hs, int nrows, int ntiles) {
  __shared__ float sW[64 * WSTR];   // 64 rows of W for this column group
  __shared__ float sA[16 * WSTR];   // 16-row input tile
  __shared__ float sB[64];          // bias slice

  const int tid   = threadIdx.x;
  const int nbase = blockIdx.y * 64;

  // Load W rows [nbase, nbase+64) into LDS (float4, padded stride)
  for (int idx = tid; idx < 64 * (D / 4); idx += 128) {
    const int r  = idx >> 5;            // 0..63
    const int c4 = (idx & 31) << 2;     // 0,4,...,124
    const float4 v = *(const float4*)(W + (size_t)(nbase + r) * D + c4);
    *(float4*)(&sW[r * WSTR + c4]) = v;
  }
  if (tid < 64) sB[tid] = bias[nbase + tid];
  __syncthreads();

  const int wave = tid >> 5;          // 0..3
  const int lane = tid & 31;
  const int hi   = lane >> 4;         // 0: K+{0,1}, 1: K+{2,3}
  const int lm   = lane & 15;

  for (int tile = blockIdx.x; tile < ntiles; tile += gridDim.x) {
    const int row0 = tile << 4;

    // Load 16 x 128 A tile into LDS
    for (int idx = tid; idx < 16 * (D / 4); idx += 128) {
      const int r   = idx >> 5;
      const int c4  = (idx & 31) << 2;
      const int row = row0 + r;
      float4 v;
      if (row < nrows) v = *(const float4*)(h + (size_t)row * D + c4);
      else { v.x = 0.f; v.y = 0.f; v.z = 0.f; v.w = 0.f; }
      *(float4*)(&sA[r * WSTR + c4]) = v;
    }
    __syncthreads();

    v8f acc = {};
    const float* __restrict__ arow = &sA[lm * WSTR];
    const float* __restrict__ brow = &sW[(wave * 16 + lm) * WSTR];
#pragma unroll
    for (int k = 0; k < D; k += 4) {
      const int kk = k + (hi << 1);
      v2f a; a.x = arow[kk]; a.y = arow[kk + 1];
      v2f b; b.x = brow[kk]; b.y = brow[kk + 1];
      acc = __builtin_amdgcn_wmma_f32_16x16x4_f32(
          /*neg_a=*/false, a, /*neg_b=*/false, b,
          /*c_mod=*/(short)0, acc, /*reuse_a=*/false, /*reuse_b=*/false);
    }

    const int n  = nbase + wave * 16 + lm;
    const float bn = sB[wave * 16 + lm];
#pragma unroll
    for (int r = 0; r < 8; ++r) {
      const int m = row0 + r + (hi << 3);
      if (m < nrows)
        hs[(size_t)m * D + n] = (acc[r] + bn) * out_norm[m];
    }
    __syncthreads();
  }
}

// ---------------------------------------------------------------------------
// agg[dst[e], :] += hs[src[e], :]   (one wave per edge; lane j handles 4 cols)
// Contiguous 512B gather + 512B of atomics per wave; hs/agg are L2-resident.
// ---------------------------------------------------------------------------
__global__ void gcn_scatter(const float* __restrict__ hs, const int* __restrict__ src,
                            const int* __restrict__ dst, float* __restrict__ agg,
                            long long E) {
  long long gid = (long long)blockIdx.x * blockDim.x + threadIdx.x;
  const long long nwork = E << 5;                  // E * 32
  if (gid >= nwork) return;
  const int e  = (int)(gid >> 5);
  const int c4 = (int)(gid & 31) << 2;
  const int s = src[e];
  const int d = dst[e];
  const float4 v = *(const float4*)(hs + (size_t)s * D + c4);
  float* o = agg + (size_t)d * D + c4;
  atomicAdd(o + 0, v.x);
  atomicAdd(o + 1, v.y);
  atomicAdd(o + 2, v.z);
  atomicAdd(o + 3, v.w);
}

// ---------------------------------------------------------------------------
// out = agg * in_norm[:,None] (+ hres) (relu?)   -- elementwise, float4
// ---------------------------------------------------------------------------
__global__ void gcn_finalize(const float* __restrict__ agg, const float* __restrict__ in_norm,
                             const float* __restrict__ hres, float* __restrict__ out,
                             int nrows, int do_relu) {
  long long gid = (long long)blockIdx.x * blockDim.x + threadIdx.x;
  const long long n4 = (long long)nrows * (D / 4);
  if (gid >= n4) return;
  const int row = (int)(gid >> 5);
  const float s = in_norm[row];
  float4 v = *(const float4*)(agg + (gid << 2));
  v.x *= s; v.y *= s; v.z *= s; v.w *= s;
  if (hres) {
    const float4 r = *(const float4*)(hres + (gid << 2));
    v.x += r.x; v.y += r.y; v.z += r.z; v.w += r.w;
  }
  if (do_relu) {
    v.x = fmaxf(v.x, 0.f); v.y = fmaxf(v.y, 0.f);
    v.z = fmaxf(v.z, 0.f); v.w = fmaxf(v.w, 0.f);
  }
  *(float4*)(out + (gid << 2)) = v;
}

// ---------------------------------------------------------------------------
// Host launch
// ---------------------------------------------------------------------------
extern "C" void kernel_launch(void* const* d_in, const int* in_sizes, int n_in,
                              void* d_out, int out_size, void* d_ws, size_t ws_size,
                              hipStream_t stream) {
  const float* feat = (const float*)d_in[0];
  const int*   src  = (const int*)d_in[1];
  const int*   dst  = (const int*)d_in[2];
  const float* Wl[3] = { (const float*)d_in[3], (const float*)d_in[5], (const float*)d_in[7] };
  const float* bl[3] = { (const float*)d_in[4], (const float*)d_in[6], (const float*)d_in[8] };

  const int N = in_sizes[0] / D;
  const int E = in_sizes[1];

  // Workspace layout
  char* ws = (char*)d_ws;
  const size_t featBytes = (size_t)N * D * sizeof(float);
  const size_t normBytes = (((size_t)N * sizeof(float)) + 255) & ~(size_t)255;
  float* hs   = (float*)ws;                                   // GEMM output buffer
  float* hbuf = (float*)(ws + featBytes);                     // layer output ping buffer
  float* onrm = (float*)(ws + 2 * featBytes);                 // out_norm
  float* inrm = (float*)(ws + 2 * featBytes + normBytes);     // in_norm
  float* agg  = (float*)d_out;                                // accumulator (reuses d_out)

  // --- degrees -> norms ---
  gcn_zero_f32<<<1024, 256, 0, stream>>>(onrm, (long long)N);
  gcn_zero_f32<<<1024, 256, 0, stream>>>(inrm, (long long)N);
  gcn_degree<<<(E + 255) / 256, 256, 0, stream>>>(src, dst, onrm, inrm, E);
  gcn_norm<<<(N + 255) / 256, 256, 0, stream>>>(onrm, N);
  gcn_norm<<<(N + 255) / 256, 256, 0, stream>>>(inrm, N);

  const int ntiles = (N + 15) / 16;
  const dim3 ggrid(1024, 2);
  const long long scatterWork = (long long)E << 5;
  const unsigned scatterBlocks = (unsigned)((scatterWork + 255) / 256);
  const long long n4 = (long long)N * (D / 4);
  const unsigned finBlocks = (unsigned)((n4 + 255) / 256);

  const float* hin = feat;
  for (int l = 0; l < 3; ++l) {
    const bool last = (l == 2);
    // hs = (h @ W^T + b) * out_norm
    gcn_gemm_wmma<<<ggrid, 128, 0, stream>>>(hin, Wl[l], bl[l], onrm, hs, N, ntiles);
    // agg = segment_sum(hs[src], dst)
    gcn_zero_f32<<<4096, 256, 0, stream>>>(agg, (long long)N * D);
    gcn_scatter<<<scatterBlocks, 256, 0, stream>>>(hs, src, dst, agg, (long long)E);
    // rst = agg * in_norm (+ h) (relu)
    float* outp = last ? (float*)d_out : hbuf;
    gcn_finalize<<<finBlocks, 256, 0, stream>>>(agg, inrm, last ? nullptr : hin,
                                                outp, N, last ? 0 : 1);
    hin = hbuf;
  }
}